// DotLayer_19894288515177
// MI455X (gfx1250) — compile-verified
//
#include <hip/hip_runtime.h>

// Row-wise cosine similarity over [65536 rows x 256 cols] f32.
// One wave32 handles 16 rows. Per 4-K step: lanes load float2 of a and b in the
// shared A/B layout of V_WMMA_F32_16X16X4_F32; the accumulated C-matrix diagonal
// is the per-row dot(a,b). ||a||^2, ||b||^2 accumulate on VALU per-lane.

typedef __attribute__((ext_vector_type(2))) float v2f;
typedef __attribute__((ext_vector_type(8))) float v8f;

#define DCOLS 256
#define KSTEPS (DCOLS / 4)   // 64 WMMA accumulation steps

__global__ __launch_bounds__(256)
void cosine_rows_wmma(const float* __restrict__ a,
                      const float* __restrict__ b,
                      float* __restrict__ out)
{
    const int lane  = threadIdx.x & 31;
    const int wave  = threadIdx.x >> 5;
    const int gwave = blockIdx.x * (blockDim.x >> 5) + wave;
    const int row0  = gwave << 4;               // 16 rows per wave

    const int r  = lane & 15;                   // row within tile
    const int cb = (lane >> 4) << 1;            // k-column base: 0 or 2

    const float* pa = a + (size_t)(row0 + r) * DCOLS + cb;
    const float* pb = b + (size_t)(row0 + r) * DCOLS + cb;

    v8f acc = {};                               // C/D accumulator (16x16 f32)
    float aa0 = 0.f, aa1 = 0.f, bb0 = 0.f, bb1 = 0.f;

    #pragma unroll 8
    for (int k = 0; k < KSTEPS; ++k) {
        // A layout 16x4 f32: v0 = {k+0 | k+2}, v1 = {k+1 | k+3}; B 4x16 is the
        // identical register image for this tile -> load once per tensor.
        v2f av = *(const v2f*)(pa + 4 * k);
        v2f bv = *(const v2f*)(pb + 4 * k);

        aa0 = fmaf(av.x, av.x, aa0);
        aa1 = fmaf(av.y, av.y, aa1);
        bb0 = fmaf(bv.x, bv.x, bb0);
        bb1 = fmaf(bv.y, bv.y, bb1);

        // C[m][n] += sum_k a[m][k] * b[n][k]  (diagonal = row-wise dot)
        acc = __builtin_amdgcn_wmma_f32_16x16x4_f32(
                  false, av, false, bv, (short)0, acc, false, false);
    }

    // Finish ||a||^2, ||b||^2: combine k-halves (lane L and L+16 share a row).
    float aa = aa0 + aa1;
    float bb = bb0 + bb1;
    aa += __shfl_xor(aa, 16, 32);
    bb += __shfl_xor(bb, 16, 32);

    // Diagonal of C: row m lives at VGPR (m%8), lane m (m<8) or lane m+16.
    // Valid lanes: 0..7 (m = lane) and 24..31 (m = lane-16); element = lane&7.
    const int e = lane & 7;
    float abv = acc[0];
    #pragma unroll
    for (int t = 1; t < 8; ++t) if (e == t) abv = acc[t];

    // Move aa/bb of row m to its diagonal lane (source lane = m = lane&15).
    float aad = __shfl(aa, r, 32);
    float bbd = __shfl(bb, r, 32);

    if (lane < 8 || lane >= 24) {
        const int m = (lane < 8) ? lane : (lane - 16);
        const float EPS = 1e-12f;
        float v = abv * rsqrtf(fmaxf(aad, EPS)) * rsqrtf(fmaxf(bbd, EPS));
        out[row0 + m] = v;
    }
}

extern "C" void kernel_launch(void* const* d_in, const int* in_sizes, int n_in,
                              void* d_out, int out_size, void* d_ws, size_t ws_size,
                              hipStream_t stream) {
    const float* a = (const float*)d_in[0];
    const float* b = (const float*)d_in[1];
    float* out = (float*)d_out;

    const int rows   = in_sizes[0] / DCOLS;   // 65536
    const int waves  = rows / 16;             // 4096 (16 rows per wave)
    const int blocks = waves / 8;             // 8 waves (256 threads) per block

    cosine_rows_wmma<<<blocks, 256, 0, stream>>>(a, b, out);
}